// ClustGeoNodeEncoder_78039555768939
// MI455X (gfx1250) — compile-verified
//
#include <hip/hip_runtime.h>
#include <math.h>

// ---------------------------------------------------------------------------
// ClustGeoNodeEncoder for MI455X (gfx1250)
//  Pass 1: LDS-privatized segment sums (10 accumulators/cluster, 160KB LDS,
//          ds_add_f32) + global_atomic_add_f32 flush
//  Per-cluster: analytic 3x3 symmetric eigensolve, B = cov/lmax, dirwt
//  Pass 2: per-cluster table staged into LDS via TENSOR_LOAD_TO_LDS (TDM,
//          s_wait_tensorcnt), sign statistic sc accumulated in LDS (144KB)
//  Finalize: signed, weighted v0
// Output row (16 floats): center[3], B[9], v0[3], count
// ---------------------------------------------------------------------------

typedef unsigned int u32x4 __attribute__((ext_vector_type(4)));
typedef int          i32x8 __attribute__((ext_vector_type(8)));
typedef int          i32x4 __attribute__((ext_vector_type(4)));

#if defined(__gfx1250__) && __has_builtin(__builtin_amdgcn_tensor_load_to_lds)
#define HAVE_TDM 1
#endif

__global__ void k_zero(float* __restrict__ p, int m) {
    int i = blockIdx.x * blockDim.x + threadIdx.x;
    if (i < m) p[i] = 0.f;
}

// Pass 1: grid-stride over voxels; privatized histogram in LDS (10*C floats)
__global__ void k_accum(const float* __restrict__ data, const int* __restrict__ ids,
                        float* __restrict__ gsums, int n, int C) {
    extern __shared__ float ls[];          // 10*C floats = 160KB for C=4096
    const int tot = 10 * C;
    for (int i = threadIdx.x; i < tot; i += blockDim.x) ls[i] = 0.f;
    __syncthreads();

    const int stride = gridDim.x * blockDim.x;
    for (int i = blockIdx.x * blockDim.x + threadIdx.x; i < n; i += stride) {
        if (i + stride < n) {
            __builtin_prefetch(data + (size_t)(i + stride) * 5, 0, 0); // global_prefetch_b8
            __builtin_prefetch(ids + (i + stride), 0, 0);
        }
        const float* row = data + (size_t)i * 5;
        float x = row[0], y = row[1], z = row[2];
        int c = ids[i];
        atomicAdd(&ls[0 * C + c], 1.f);        // ds_add_f32 (no-return)
        atomicAdd(&ls[1 * C + c], x);
        atomicAdd(&ls[2 * C + c], y);
        atomicAdd(&ls[3 * C + c], z);
        atomicAdd(&ls[4 * C + c], x * x);
        atomicAdd(&ls[5 * C + c], x * y);
        atomicAdd(&ls[6 * C + c], x * z);
        atomicAdd(&ls[7 * C + c], y * y);
        atomicAdd(&ls[8 * C + c], y * z);
        atomicAdd(&ls[9 * C + c], z * z);
    }
    __syncthreads();
    for (int i = threadIdx.x; i < tot; i += blockDim.x)
        atomicAdd(&gsums[i], ls[i]);           // global_atomic_add_f32
}

__device__ inline void cross3(float ax, float ay, float az,
                              float bx, float by, float bz,
                              float& cx, float& cy, float& cz) {
    cx = ay * bz - az * by;
    cy = az * bx - ax * bz;
    cz = ax * by - ay * bx;
}

// Per-cluster feature construction (one thread per cluster)
__global__ void k_cluster(const float* __restrict__ gsums,
                          float* __restrict__ out,        // C x 16
                          float* __restrict__ cinfo,      // C x 8: cx,cy,cz,vx,vy,vz,dirwt,small
                          int C) {
    int c = blockIdx.x * blockDim.x + threadIdx.x;
    if (c >= C) return;

    float cnt = gsums[0 * C + c];
    float sx  = gsums[1 * C + c], sy  = gsums[2 * C + c], sz  = gsums[3 * C + c];
    float sxx = gsums[4 * C + c], sxy = gsums[5 * C + c], sxz = gsums[6 * C + c];
    float syy = gsums[7 * C + c], syz = gsums[8 * C + c], szz = gsums[9 * C + c];

    float safe = fmaxf(cnt, 1.f);
    float inv  = 1.f / safe;
    float cx = sx * inv, cy = sy * inv, cz = sz * inv;

    // cov = S2 - S*S^T / count  (exact for cnt>=1; zero for empty clusters)
    float axx = sxx - sx * cx;
    float axy = sxy - sx * cy;
    float axz = sxz - sx * cz;
    float ayy = syy - sy * cy;
    float ayz = syz - sy * cz;
    float azz = szz - sz * cz;

    // --- analytic 3x3 symmetric eigensolve (trig / Smith method) ---
    float lmax, lmid, vx, vy, vz;
    float p1 = axy * axy + axz * axz + ayz * ayz;
    if (p1 == 0.f) {
        // exactly diagonal (includes the exact-zero matrix of empty clusters)
        float l0 = axx, l1 = ayy, l2 = azz;
        lmax = fmaxf(l0, fmaxf(l1, l2));
        float lmin = fminf(l0, fminf(l1, l2));
        lmid = l0 + l1 + l2 - lmax - lmin;
        vx = (lmax == l0) ? 1.f : 0.f;
        vy = (lmax == l0) ? 0.f : ((lmax == l1) ? 1.f : 0.f);
        vz = (vx == 0.f && vy == 0.f) ? 1.f : 0.f;
    } else {
        float q = (axx + ayy + azz) * (1.f / 3.f);
        float bxx = axx - q, byy = ayy - q, bzz = azz - q;
        float p2 = bxx * bxx + byy * byy + bzz * bzz + 2.f * p1;
        float p = sqrtf(p2 * (1.f / 6.f));
        float invp = 1.f / p;
        float detB = bxx * (byy * bzz - ayz * ayz)
                   - axy * (axy * bzz - ayz * axz)
                   + axz * (axy * ayz - byy * axz);
        float r = 0.5f * detB * invp * invp * invp;
        r = fminf(1.f, fmaxf(-1.f, r));
        float phi = acosf(r) * (1.f / 3.f);
        lmax = q + 2.f * p * cosf(phi);
        float lmin = q + 2.f * p * cosf(phi + 2.0943951023931953f); // +2pi/3
        lmid = 3.f * q - lmax - lmin;

        // eigenvector of lmax: cross products of rows of (A - lmax*I)
        float m00 = axx - lmax, m11 = ayy - lmax, m22 = azz - lmax;
        float c0x, c0y, c0z, c1x, c1y, c1z, c2x, c2y, c2z;
        cross3(m00, axy, axz,  axy, m11, ayz,  c0x, c0y, c0z);
        cross3(axy, m11, ayz,  axz, ayz, m22,  c1x, c1y, c1z);
        cross3(axz, ayz, m22,  m00, axy, axz,  c2x, c2y, c2z);
        float n0 = c0x * c0x + c0y * c0y + c0z * c0z;
        float n1 = c1x * c1x + c1y * c1y + c1z * c1z;
        float n2 = c2x * c2x + c2y * c2y + c2z * c2z;
        float best = n0; vx = c0x; vy = c0y; vz = c0z;
        if (n1 > best) { best = n1; vx = c1x; vy = c1y; vz = c1z; }
        if (n2 > best) { best = n2; vx = c2x; vy = c2y; vz = c2z; }
        if (best > 0.f) {
            float rn = rsqrtf(best);
            vx *= rn; vy *= rn; vz *= rn;
        } else {
            vx = vy = vz = 0.f;  // degenerate top eigenspace -> dirwt ~ 0 anyway
        }
    }

    bool  small  = cnt < 2.f;
    float denom  = (lmax == 0.f) ? 1.f : lmax;
    float invd   = 1.f / denom;
    float dirwt  = (lmax == 0.f) ? 0.f : (1.f - lmid / lmax);
    float bscale = small ? 0.f : invd;   // DELTA==0: B = cov/denom, zero if small

    float* o = out + (size_t)c * 16;
    o[0] = cx; o[1] = cy; o[2] = cz;
    o[3]  = axx * bscale; o[4]  = axy * bscale; o[5]  = axz * bscale;
    o[6]  = axy * bscale; o[7]  = ayy * bscale; o[8]  = ayz * bscale;
    o[9]  = axz * bscale; o[10] = ayz * bscale; o[11] = azz * bscale;
    o[15] = cnt;

    float* ci = cinfo + (size_t)c * 8;
    ci[0] = cx; ci[1] = cy; ci[2] = cz;
    ci[3] = vx; ci[4] = vy; ci[5] = vz;
    ci[6] = dirwt;
    ci[7] = small ? 1.f : 0.f;
}

// Pass 2: sc = segment_sum(x0 * ||xp0||); cinfo (C x 8 floats, 128KB) staged
// into LDS by the Tensor Data Mover; sc privatized in LDS.
__global__ void k_sign(const float* __restrict__ data, const int* __restrict__ ids,
                       const float* __restrict__ cinfo, float* __restrict__ gsc,
                       int n, int C) {
    extern __shared__ float ls[];          // 8*C (cinfo copy) + C (sc) = 144KB
    float* lcv = ls;                       // dynamic LDS base -> lds_addr 0
    float* lsc = ls + 8 * C;

    const int elems = 8 * C;               // dwords to stage (32768 @ C=4096)

#ifdef HAVE_TDM
    if (threadIdx.x == 0) {
        // D# group0: count=1 | lds_addr=0 | 57-bit global addr | type=2
        unsigned long long ga = (unsigned long long)(uintptr_t)cinfo;
        u32x4 g0;
        g0.x = 1u;                                      // count=1, user mode
        g0.y = 0u;                                      // lds_addr (bytes): LDS base
        g0.z = (unsigned)ga;                            // global_addr[31:0]
        g0.w = (unsigned)((ga >> 32) & 0x1FFFFFFu) | (2u << 30); // addr[56:32] | type=2
        // D# group1: 1 row of `elems` dwords
        i32x8 g1;
        g1[0] = (2 << 16);                              // wg_mask=0, data_size=4B
        g1[1] = (int)(((unsigned)elems & 0xFFFFu) << 16);        // tensor_dim0[15:0]
        g1[2] = (int)((((unsigned)elems >> 16) & 0xFFFFu) | (1u << 16)); // dim0 hi | tensor_dim1=1
        g1[3] = (int)(((unsigned)elems & 0xFFFFu) << 16);        // tile_dim0
        g1[4] = 1;                                      // tile_dim1=1, tile_dim2 unused
        g1[5] = elems;                                  // tensor_dim0_stride[31:0]
        g1[6] = (int)(((unsigned)elems & 0xFFFFu) << 16);        // dim1_stride[15:0]
        g1[7] = (int)(((unsigned)elems >> 16) & 0xFFFFu);        // dim1_stride[47:16]
        i32x4 gz = {0, 0, 0, 0};                        // groups 2/3 unused (<=2D)
#if __clang_major__ >= 23
        i32x8 gz8 = {0, 0, 0, 0, 0, 0, 0, 0};
        __builtin_amdgcn_tensor_load_to_lds(g0, g1, gz, gz, gz8, 0);
#else
        __builtin_amdgcn_tensor_load_to_lds(g0, g1, gz, gz, 0);
#endif
        __builtin_amdgcn_s_wait_tensorcnt(0);           // s_wait_tensorcnt 0
    }
    for (int c = threadIdx.x; c < C; c += blockDim.x) lsc[c] = 0.f;
#else
    for (int c = threadIdx.x; c < C; c += blockDim.x) {
        const float* ci = cinfo + (size_t)c * 8;
        lcv[c * 8 + 0] = ci[0]; lcv[c * 8 + 1] = ci[1]; lcv[c * 8 + 2] = ci[2];
        lcv[c * 8 + 3] = ci[3]; lcv[c * 8 + 4] = ci[4]; lcv[c * 8 + 5] = ci[5];
        lsc[c] = 0.f;
    }
#endif
    (void)elems;
    __syncthreads();

    const int stride = gridDim.x * blockDim.x;
    for (int i = blockIdx.x * blockDim.x + threadIdx.x; i < n; i += stride) {
        if (i + stride < n) {
            __builtin_prefetch(data + (size_t)(i + stride) * 5, 0, 0);
            __builtin_prefetch(ids + (i + stride), 0, 0);
        }
        const float* row = data + (size_t)i * 5;
        float x = row[0], y = row[1], z = row[2];
        int c = ids[i];
        const float* cv = lcv + c * 8;
        float dx = x - cv[0], dy = y - cv[1], dz = z - cv[2];
        float vx = cv[3], vy = cv[4], vz = cv[5];
        float x0 = dx * vx + dy * vy + dz * vz;
        float px = dx - x0 * vx, py = dy - x0 * vy, pz = dz - x0 * vz;
        float np = sqrtf(px * px + py * py + pz * pz);
        atomicAdd(&lsc[c], x0 * np);       // ds_add_f32
    }
    __syncthreads();
    for (int c = threadIdx.x; c < C; c += blockDim.x)
        atomicAdd(&gsc[c], lsc[c]);
}

__global__ void k_final(const float* __restrict__ gsc, const float* __restrict__ cinfo,
                        float* __restrict__ out, int C) {
    int c = blockIdx.x * blockDim.x + threadIdx.x;
    if (c >= C) return;
    const float* ci = cinfo + (size_t)c * 8;
    float sc  = gsc[c];
    float sgn = (sc < 0.f) ? -1.f : 1.f;
    float s   = (ci[7] != 0.f) ? 0.f : sgn * ci[6];   // zero if small, else +/- dirwt
    float* o = out + (size_t)c * 16;
    o[12] = s * ci[3];
    o[13] = s * ci[4];
    o[14] = s * ci[5];
}

extern "C" void kernel_launch(void* const* d_in, const int* in_sizes, int n_in,
                              void* d_out, int out_size, void* d_ws, size_t ws_size,
                              hipStream_t stream) {
    (void)n_in; (void)ws_size;
    const float* data = (const float*)d_in[0];
    const int*   ids  = (const int*)d_in[1];
    const int n = in_sizes[0] / 5;
    const int C = out_size / 16;          // 16 features per cluster

    float* ws    = (float*)d_ws;
    float* gsums = ws;                    // 10*C
    float* gsc   = ws + (size_t)10 * C;   // C
    float* cinfo = ws + (size_t)11 * C;   // 8*C
    float* out   = (float*)d_out;

    const int  T      = 256;              // 8 wave32 waves per block
    const int  NBLK   = 256;              // privatized blocks (flush = NBLK*10*C atomics)
    const size_t ldsA = (size_t)10 * C * sizeof(float);  // 160KB @ C=4096 (needs 320KB WGP LDS)
    const size_t ldsS = (size_t)9  * C * sizeof(float);  // 144KB

    k_zero   <<<(11 * C + T - 1) / T, T, 0,    stream>>>(ws, 11 * C);
    k_accum  <<<NBLK,               T, ldsA, stream>>>(data, ids, gsums, n, C);
    k_cluster<<<(C + T - 1) / T,    T, 0,    stream>>>(gsums, out, cinfo, C);
    k_sign   <<<NBLK,               T, ldsS, stream>>>(data, ids, cinfo, gsc, n, C);
    k_final  <<<(C + T - 1) / T,    T, 0,    stream>>>(gsc, cinfo, out, C);
}